// MobiusRNN_37950331027528
// MI455X (gfx1250) — compile-verified
//
#include <hip/hip_runtime.h>

// ---------------------------------------------------------------------------
// Mobius RNN on the Poincare ball for MI455X (gfx1250, wave32, WMMA).
//   Kernel 1 (parallel):  ux[b,s,:] = mobius_matvec(u, x[b,s,:])  (B*S rows)
//   Kernel 2 (recurrent): 16 batch rows per WG; per step a 16x256x256 GEMM
//                         via v_wmma_f32_16x16x4_f32 with w held in VGPRs,
//                         then row-wise Mobius projection + two Mobius adds.
// A-fragment LDS loads are software-pipelined (2x8 double buffer) and the
// pipeline is pinned with sched_barrier(0) so the scheduler cannot sink the
// prefetch loads back to their consumers (which would serialize the WMMA
// chain on s_wait_dscnt 0 as observed in round 2's asm).
// ---------------------------------------------------------------------------

typedef __attribute__((ext_vector_type(2))) float v2f;
typedef __attribute__((ext_vector_type(8))) float v8f;

#define MOB_EPS   1e-15f
#define MOB_BOUND (1.0f - 1e-5f)

#if defined(__has_builtin)
#if __has_builtin(__builtin_amdgcn_sched_barrier)
#define SCHED_FENCE() __builtin_amdgcn_sched_barrier(0)
#endif
#endif
#ifndef SCHED_FENCE
#define SCHED_FENCE()
#endif

__device__ __forceinline__ float warp_sum32(float v) {
#pragma unroll
  for (int m = 16; m >= 1; m >>= 1) v += __shfl_xor(v, m, 32);
  return v;
}

__device__ __forceinline__ float artanh_clamped(float x) {
  x = fminf(fmaxf(x, -MOB_BOUND), MOB_BOUND);
  return 0.5f * logf((1.0f + x) / (1.0f - x));
}

// Load the B-fragments (16 rows of W == 16 cols of W^T) for one wave's N-tile
// across all K=256 (64 wmma steps). Kept resident in 128 VGPRs.
// B[k][n] = W[n*256 + k].  4x16 f32 B layout: lanes 0-15 -> K={k0,k0+1},
// lanes 16-31 -> K={k0+2,k0+3}, N = lane&15.
__device__ __forceinline__ void load_bfrag(const float* __restrict__ W,
                                           int n0, v2f* bfrag) {
  const int lane = threadIdx.x & 31;
  const int n    = n0 + (lane & 15);
  const int khi  = (lane >> 4) << 1;
  const float* base = W + n * 256 + khi;
#pragma unroll
  for (int kk = 0; kk < 64; ++kk)
    bfrag[kk] = *(const v2f*)(base + kk * 4);
}

// D(16x16) = A(16x256) * B(256x16), A rows in LDS (row stride 256 floats).
// 16x4 f32 A layout: lanes 0-15 -> K={k0,k0+1}, lanes 16-31 -> K={k0+2,k0+3},
// M = lane&15.  Chunk c+1's A loads are issued (and fenced) ahead of chunk
// c's 8 WMMAs so LDS latency overlaps the accumulate chain.
__device__ __forceinline__ v8f gemm_tile(const float* __restrict__ ldsA,
                                         const v2f* bfrag) {
  const int lane = threadIdx.x & 31;
  const float* ap = ldsA + (lane & 15) * 256 + ((lane >> 4) << 1);
  v2f a0[8], a1[8];
  v8f acc = {};
#pragma unroll
  for (int i = 0; i < 8; ++i) a0[i] = *(const v2f*)(ap + i * 4);
  SCHED_FENCE();
#pragma unroll
  for (int c = 0; c < 8; ++c) {
    const v2f* cur = (c & 1) ? a1 : a0;
    v2f*       nxt = (c & 1) ? a0 : a1;
    if (c < 7) {
#pragma unroll
      for (int i = 0; i < 8; ++i)
        nxt[i] = *(const v2f*)(ap + ((c + 1) * 8 + i) * 4);
    }
    SCHED_FENCE();   // prefetch for c+1 stays above the WMMAs of chunk c
#pragma unroll
    for (int i = 0; i < 8; ++i)
      acc = __builtin_amdgcn_wmma_f32_16x16x4_f32(
          /*neg_a=*/false, cur[i], /*neg_b=*/false, bfrag[c * 8 + i],
          /*c_mod=*/(short)0, acc, /*reuse_a=*/false, /*reuse_b=*/false);
    SCHED_FENCE();
  }
  return acc;
}

// Scatter a 16x16 f32 D tile to LDS (row stride 256). C/D layout:
// VGPR j: lanes 0-15 -> M=j, lanes 16-31 -> M=8+j, N = lane&15.
__device__ __forceinline__ void store_tile(float* ldsD, int n0, v8f acc) {
  const int lane  = threadIdx.x & 31;
  const int n     = n0 + (lane & 15);
  const int rbase = (lane >> 4) << 3;
#pragma unroll
  for (int j = 0; j < 8; ++j) ldsD[(rbase + j) * 256 + n] = acc[j];
}

// ---------------------------------------------------------------------------
// Kernel 1: ux = mobius_matvec(u, x) for all B*S rows (grid-stride M-tiles).
// ---------------------------------------------------------------------------
__global__ __launch_bounds__(512) void
mobius_ux_kernel(const float* __restrict__ x, const float* __restrict__ u,
                 float* __restrict__ ux, int num_tiles) {
  __shared__ float lds_x[16 * 256];
  __shared__ float lds_mx[16 * 256];
  const int tid  = threadIdx.x;
  const int wv   = tid >> 5;   // wave id = N-tile index = elementwise row
  const int lane = tid & 31;

  v2f bfrag[64];
  load_bfrag(u, wv * 16, bfrag);   // u stays in registers for all tiles

  for (int t = blockIdx.x; t < num_tiles; t += gridDim.x) {
    const long long r0 = (long long)t * 16;
    // Stage 16 input rows (16KB) into LDS, coalesced; x is read-once -> NT.
#pragma unroll
    for (int j = 0; j < 8; ++j) {
      const int idx = tid + 512 * j;
      lds_x[idx] = __builtin_nontemporal_load(x + r0 * 256 + idx);
    }
    __syncthreads();

    v8f acc = gemm_tile(lds_x, bfrag);      // mx = x @ u.T (one N-tile)
    store_tile(lds_mx, wv * 16, acc);
    __syncthreads();

    // Row-wise Mobius projection: wave wv owns row wv.
    const float* mxr = lds_mx + wv * 256;
    const float* xr  = lds_x  + wv * 256;
    float m[8], x2p = 0.f, m2p = 0.f;
#pragma unroll
    for (int i = 0; i < 8; ++i) {
      const float xv = xr[lane + 32 * i];
      m[i] = mxr[lane + 32 * i];
      x2p += xv * xv;
      m2p += m[i] * m[i];
    }
    const float xn = sqrtf(fmaxf(warp_sum32(x2p), MOB_EPS));
    const float mn = sqrtf(fmaxf(warp_sum32(m2p), MOB_EPS));
    const float sc = tanhf(mn / xn * artanh_clamped(xn)) / mn;

    // Regular-temporal stores: kernel 2 re-reads ux, keep it hot in L2.
    float* outr = ux + (r0 + wv) * 256;
#pragma unroll
    for (int i = 0; i < 8; ++i) outr[lane + 32 * i] = sc * m[i];
    __syncthreads();
  }
}

// ---------------------------------------------------------------------------
// Kernel 2: the recurrence. One WG = 16 batch rows; 512 sequential steps.
// ---------------------------------------------------------------------------
__global__ __launch_bounds__(512) void
mobius_rnn_kernel(const float* __restrict__ ux, const float* __restrict__ w,
                  const float* __restrict__ bias, float* __restrict__ out) {
  __shared__ float lds_h[16 * 256];
  __shared__ float lds_wh[16 * 256];
  const int tid  = threadIdx.x;
  const int wv   = tid >> 5;
  const int lane = tid & 31;
  const int b0   = blockIdx.x * 16;

  v2f bfrag[64];
  load_bfrag(w, wv * 16, bfrag);   // w resident in VGPRs for all 512 steps

  // Bias row in registers + its (constant) squared norm.
  float bv[8];
  float b2p = 0.f;
#pragma unroll
  for (int i = 0; i < 8; ++i) {
    bv[i] = bias[lane + 32 * i];
    b2p += bv[i] * bv[i];
  }
  const float b2 = warp_sum32(b2p);

  // h0 = 0
#pragma unroll
  for (int j = 0; j < 8; ++j) lds_h[tid + 512 * j] = 0.0f;
  __syncthreads();

  for (int s = 0; s < 512; ++s) {
    // wh_raw = h @ w.T  (pipelined WMMA accumulate chain)
    v8f acc = gemm_tile(lds_h, bfrag);
    store_tile(lds_wh, wv * 16, acc);
    __syncthreads();

    // Wave wv handles batch row b0+wv; lane holds cols {lane + 32*i}.
    const float* hr  = lds_h  + wv * 256;
    const float* mr  = lds_wh + wv * 256;
    const float* uxr = ux + ((long long)(b0 + wv) * 512 + s) * 256;

    float m[8], h2p = 0.f, m2p = 0.f;
#pragma unroll
    for (int i = 0; i < 8; ++i) {
      const float hv = hr[lane + 32 * i];
      m[i] = mr[lane + 32 * i];
      h2p += hv * hv;
      m2p += m[i] * m[i];
    }
    const float m2 = warp_sum32(m2p);
    const float hn = sqrtf(fmaxf(warp_sum32(h2p), MOB_EPS));
    const float mn = sqrtf(fmaxf(m2, MOB_EPS));
    const float tv  = tanhf(mn / hn * artanh_clamped(hn));
    const float wsc = tv / mn;          // wh = wsc * m
    const float x2  = tv * tv;          // ||wh||^2

    // z1 = mobius_add(wh, ux)
    float uxv[8], y2p = 0.f, xyp = 0.f;
#pragma unroll
    for (int i = 0; i < 8; ++i) {
      uxv[i] = uxr[lane + 32 * i];
      y2p += uxv[i] * uxv[i];
      xyp += (wsc * m[i]) * uxv[i];
    }
    const float y2 = warp_sum32(y2p);
    const float xy = warp_sum32(xyp);
    const float ca = 1.f + 2.f * xy + y2;
    const float cb = 1.f - x2;
    const float cd = 1.f / fmaxf(1.f + 2.f * xy + x2 * y2, MOB_EPS);

    // z2 = mobius_add(z1, b)
    float z1[8], zzp = 0.f, zbp = 0.f;
#pragma unroll
    for (int i = 0; i < 8; ++i) {
      z1[i] = (ca * (wsc * m[i]) + cb * uxv[i]) * cd;
      zzp += z1[i] * z1[i];
      zbp += z1[i] * bv[i];
    }
    const float zz  = warp_sum32(zzp);
    const float zb  = warp_sum32(zbp);
    const float ca2 = 1.f + 2.f * zb + b2;
    const float cb2 = 1.f - zz;
    const float cd2 = 1.f / fmaxf(1.f + 2.f * zb + zz * b2, MOB_EPS);

    float* outr = out + ((long long)(b0 + wv) * 512 + s) * 256;
    float* hw   = lds_h + wv * 256;
#pragma unroll
    for (int i = 0; i < 8; ++i) {
      const float z = (ca2 * z1[i] + cb2 * bv[i]) * cd2;
      __builtin_nontemporal_store(z, outr + lane + 32 * i);  // never re-read
      hw[lane + 32 * i] = z;                                 // carry to s+1
    }
    __syncthreads();
  }
}

// ---------------------------------------------------------------------------
extern "C" void kernel_launch(void* const* d_in, const int* in_sizes, int n_in,
                              void* d_out, int out_size, void* d_ws,
                              size_t ws_size, hipStream_t stream) {
  (void)in_sizes; (void)n_in; (void)out_size; (void)ws_size;
  const float* x    = (const float*)d_in[0];   // (256, 512, 256)
  const float* w    = (const float*)d_in[1];   // (256, 256)
  const float* u    = (const float*)d_in[2];   // (256, 256)
  const float* bias = (const float*)d_in[3];   // (256,)
  float* out = (float*)d_out;                  // (256, 512, 256)
  float* ux  = (float*)d_ws;                   // B*S*H floats = 128 MB scratch

  const int num_tiles = (256 * 512) / 16;      // 8192 M-tiles of 16 rows
  mobius_ux_kernel<<<512, 512, 0, stream>>>(x, u, ux, num_tiles);
  mobius_rnn_kernel<<<16, 512, 0, stream>>>(ux, w, bias, out);
}